// PushingDynamicsGP_38689065402654
// MI455X (gfx1250) — compile-verified
//
#include <hip/hip_runtime.h>

// ---------------------------------------------------------------------------
// PushingDynamicsGP on MI455X (gfx1250).
//   1. An = -(Kxx + noise I)                  (2 x N x N, fp32)
//   2. S_t = max row sum  -> X0 = I/S        (Gershgorin => NS converges)
//   3. Newton-Schulz X <- X*(An*X) + 2X, 16 iters, LDS-tiled fp32 WMMA GEMMs
//      with double-buffered ASYNC global->LDS staging (ASYNCcnt path)
//      (all four NxN fp32 matrices = 134MB stay resident in the 192MB L2)
//   4. alpha = X*resid, ms = X*train_y (GEMV)
//   5. per 32-row m-tile: Kstar rows -> 257KB LDS (320KB/WGP), then
//      q = rowdot(K*X, K) via WMMA with A from LDS; each X load feeds 2 WMMAs
//   6. finalize pred_mu, pred_sigma = diag(var) + J Sigma J^T
// ---------------------------------------------------------------------------

#define Nn 2048
#define Mm 8192
#define NS_ITERS 16          // even -> ping-pong ends in X buffer

typedef __attribute__((ext_vector_type(2))) float v2f;
typedef __attribute__((ext_vector_type(8))) float v8f;
typedef __attribute__((ext_vector_type(4))) int   v4i;

__device__ __forceinline__ v8f wmma_f32x4(v2f a, v2f b, v8f c) {
  // D = A(16x4) * B(4x16) + C(16x16), fp32, wave32
  return __builtin_amdgcn_wmma_f32_16x16x4_f32(
      /*neg_a=*/false, a, /*neg_b=*/false, b,
      /*c_mod=*/(short)0, c, /*reuse_a=*/false, /*reuse_b=*/false);
}

#if defined(__has_builtin)
#if __has_builtin(__builtin_amdgcn_global_load_async_to_lds_b128) && \
    __has_builtin(__builtin_amdgcn_s_wait_asynccnt)
#define HAVE_ASYNC_COPY 1
#endif
#endif

#ifdef HAVE_ASYNC_COPY
// 16B global -> LDS, tracked by ASYNCcnt, completes in issue order.
// Builtin signature (probe-confirmed): (v4i __device__*, v4i __shared__*, imm, imm)
__device__ __forceinline__ void async_copy16(const float* gsrc, float* ldst) {
  void* gv = (void*)gsrc;
  void* lv = (void*)ldst;
  __builtin_amdgcn_global_load_async_to_lds_b128(
      (__attribute__((address_space(1))) v4i*)gv,
      (__attribute__((address_space(3))) v4i*)lv, 0, 0);
}
#endif

// ---------------------------------------------------------------------------
// An[t][i][j] = -( os_t * exp(-0.5 * sum_d W_td (tx_i - tx_j)^2 ) + (i==j)*noise )
// Negated so Newton-Schulz needs no operand negation.
// ---------------------------------------------------------------------------
__global__ __launch_bounds__(256)
void build_An(const float* __restrict__ train_x,
              const float* __restrict__ lengthscale,
              const float* __restrict__ outputscale,
              const float* __restrict__ noise,
              float* __restrict__ An)
{
  int t = blockIdx.z;
  int i = blockIdx.y * 16 + threadIdx.y;
  int j = blockIdx.x * 16 + threadIdx.x;
  float s = 0.f;
#pragma unroll
  for (int d = 0; d < 5; ++d) {
    float ls = lengthscale[t * 5 + d];
    float W  = 1.0f / (ls * ls);
    float df = train_x[i * 5 + d] - train_x[j * 5 + d];
    s += W * df * df;
  }
  float v = outputscale[t] * __expf(-0.5f * s) + ((i == j) ? noise[0] : 0.f);
  An[((size_t)t * Nn + i) * Nn + j] = -v;
}

__global__ void zeroS(int* __restrict__ Sb) {
  if (threadIdx.x < 2) Sb[threadIdx.x] = 0;
}

// max |row sum| per task (positive floats: int-bit atomicMax is order-preserving)
__global__ __launch_bounds__(256)
void rowscale(const float* __restrict__ An, int* __restrict__ Sb)
{
  int t   = blockIdx.y;
  int row = blockIdx.x * blockDim.x + threadIdx.x;
  const float* Ar = An + ((size_t)t * Nn + row) * Nn;
  float s = 0.f;
  for (int j = 0; j < Nn; ++j) s += fabsf(Ar[j]);
  atomicMax(&Sb[t], __float_as_int(s));
}

__global__ __launch_bounds__(256)
void initX(const int* __restrict__ Sb, float* __restrict__ X)
{
  int t = blockIdx.z;
  int i = blockIdx.y * 16 + threadIdx.y;
  int j = blockIdx.x * 16 + threadIdx.x;
  float S = __int_as_float(Sb[t]);
  X[((size_t)t * Nn + i) * Nn + j] = (i == j) ? (1.0f / S) : 0.0f;
}

// ---------------------------------------------------------------------------
// LDS-tiled GEMM:  D = A*B + cscale*C   (per-task via blockIdx.z)
// 128x128 macro-tile per 256-thread block (8 waves, each 32x64 = 8 WMMA tiles),
// k-block = 16, double-buffered in LDS. On toolchains with the gfx1250 async
// builtins, staging uses GLOBAL_LOAD_ASYNC_TO_LDS_B128 (ASYNCcnt, in-order):
// issue next k-block into buf^1, s_wait_asynccnt 4 => current buf complete.
// ---------------------------------------------------------------------------
#define ASTR 20    // A tile LDS row stride (floats): 16B-aligned rows, no bank conflicts
#define BSTR 132   // B tile LDS row stride (floats): 16B-aligned rows

__global__ __launch_bounds__(256)
void gemm_ns(const float* __restrict__ Ag, const float* __restrict__ Bg,
             const float* __restrict__ Cg, float* __restrict__ Dg, float cscale)
{
  __shared__ float Ash[2][128 * ASTR];   // 2 x 10.0 KB
  __shared__ float Bsh[2][16 * BSTR];    // 2 x  8.25 KB

  const size_t mo = (size_t)blockIdx.z * (size_t)Nn * Nn;
  const float* A  = Ag + mo;
  const float* B  = Bg + mo;
  const float* C  = Cg + mo;
  float*       Dm = Dg + mo;

  const int tid  = threadIdx.x;
  const int lane = tid & 31, wave = tid >> 5;
  const int r0b  = blockIdx.y * 128;
  const int c0b  = blockIdx.x * 128;
  const int wr   = wave >> 1;          // 0..3 -> 32-row group
  const int wc   = wave & 1;           // 0..1 -> 64-col group
  const int cl   = lane & 15, half = lane >> 4, kk0 = half * 2;

  v8f acc[2][4];
  if (cscale != 0.0f) {                // uniform branch: EXEC stays all-ones
#pragma unroll
    for (int i = 0; i < 2; ++i)
#pragma unroll
      for (int j = 0; j < 4; ++j)
#pragma unroll
        for (int v = 0; v < 8; ++v)
          acc[i][j][v] = cscale *
              C[(size_t)(r0b + wr * 32 + i * 16 + v + 8 * half) * Nn +
                c0b + wc * 64 + j * 16 + cl];
  } else {
#pragma unroll
    for (int i = 0; i < 2; ++i)
#pragma unroll
      for (int j = 0; j < 4; ++j)
        acc[i][j] = (v8f){};
  }

  // 8 WMMAs per two LDS v2f A-reads and four v2f B-reads (16x operand reuse)
  auto compute_kblock = [&](const float* __restrict__ As,
                            const float* __restrict__ Bs) {
#pragma unroll
    for (int kk = 0; kk < 16; kk += 4) {
      v2f a0, a1;
      a0.x = As[(wr * 32 +      cl) * ASTR + kk + kk0];
      a0.y = As[(wr * 32 +      cl) * ASTR + kk + kk0 + 1];
      a1.x = As[(wr * 32 + 16 + cl) * ASTR + kk + kk0];
      a1.y = As[(wr * 32 + 16 + cl) * ASTR + kk + kk0 + 1];
#pragma unroll
      for (int j = 0; j < 4; ++j) {
        int col = wc * 64 + j * 16 + cl;
        v2f b;
        b.x = Bs[(kk + kk0)     * BSTR + col];
        b.y = Bs[(kk + kk0 + 1) * BSTR + col];
        acc[0][j] = wmma_f32x4(a0, b, acc[0][j]);
        acc[1][j] = wmma_f32x4(a1, b, acc[1][j]);
      }
    }
  };

#ifdef HAVE_ASYNC_COPY
  // async double-buffered pipeline: 4 in-order async b128 copies per thread
  auto stage_async = [&](int kb, int buf) {
#pragma unroll
    for (int q = 0; q < 2; ++q) {
      int f = tid + q * 256, ar = f >> 2, ac = (f & 3) * 4;
      async_copy16(&A[(size_t)(r0b + ar) * Nn + kb + ac], &Ash[buf][ar * ASTR + ac]);
    }
#pragma unroll
    for (int q = 0; q < 2; ++q) {
      int f = tid + q * 256, br = f >> 5, bc = (f & 31) * 4;
      async_copy16(&B[(size_t)(kb + br) * Nn + c0b + bc], &Bsh[buf][br * BSTR + bc]);
    }
  };

  stage_async(0, 0);
  for (int kb = 0; kb < Nn; kb += 16) {
    int buf = (kb >> 4) & 1;
    if (kb + 16 < Nn) {                              // uniform branch
      stage_async(kb + 16, buf ^ 1);                 // overlap next copy
      __builtin_amdgcn_s_wait_asynccnt(4);           // current 4 done (in-order)
    } else {
      __builtin_amdgcn_s_wait_asynccnt(0);
    }
    __syncthreads();                                 // all waves' copies visible
    compute_kblock(Ash[buf], Bsh[buf]);
    __syncthreads();                                 // done reading buf
  }
#else
  // synchronous fallback: load->VGPR->ds_store staging (+ next-k prefetch)
  for (int kb = 0; kb < Nn; kb += 16) {
    int buf = (kb >> 4) & 1;
    if (kb + 16 < Nn) {
      __builtin_prefetch(&A[(size_t)(r0b + (tid >> 1)) * Nn + kb + 16 + (tid & 1) * 8], 0, 3);
      __builtin_prefetch(&B[(size_t)(kb + 16 + (tid >> 5)) * Nn + c0b + (tid & 31) * 4], 0, 3);
    }
#pragma unroll
    for (int q = 0; q < 2; ++q) {
      int f  = tid + q * 256, ar = f >> 2, ac = (f & 3) * 4;
      float4 av = *(const float4*)&A[(size_t)(r0b + ar) * Nn + kb + ac];
      float* dst = &Ash[buf][ar * ASTR + ac];
      dst[0] = av.x; dst[1] = av.y; dst[2] = av.z; dst[3] = av.w;
    }
#pragma unroll
    for (int q = 0; q < 2; ++q) {
      int f  = tid + q * 256, br = f >> 5, bc = (f & 31) * 4;
      float4 bv = *(const float4*)&B[(size_t)(kb + br) * Nn + c0b + bc];
      float* dst = &Bsh[buf][br * BSTR + bc];
      dst[0] = bv.x; dst[1] = bv.y; dst[2] = bv.z; dst[3] = bv.w;
    }
    __syncthreads();
    compute_kblock(Ash[buf], Bsh[buf]);
    __syncthreads();
  }
#endif

#pragma unroll
  for (int i = 0; i < 2; ++i)
#pragma unroll
    for (int j = 0; j < 4; ++j)
#pragma unroll
      for (int v = 0; v < 8; ++v)
        Dm[(size_t)(r0b + wr * 32 + i * 16 + v + 8 * half) * Nn +
           c0b + wc * 64 + j * 16 + cl] = acc[i][j][v];
}

// ---------------------------------------------------------------------------
// alpha[t] = X * (train_y[:,t] - train_x[:,t]);  ms[t] = X * train_y[:,t]
// ---------------------------------------------------------------------------
__global__ __launch_bounds__(256)
void solve_vecs(const float* __restrict__ X,
                const float* __restrict__ train_x,
                const float* __restrict__ train_y,
                float* __restrict__ alpha, float* __restrict__ msv)
{
  int t    = blockIdx.y;
  int row  = blockIdx.x * 8 + (threadIdx.x >> 5);
  int lane = threadIdx.x & 31;
  const float* Xr = X + ((size_t)t * Nn + row) * Nn;
  float accA = 0.f, accM = 0.f;
  for (int j = lane; j < Nn; j += 32) {
    float xv = Xr[j];
    float ty = train_y[j * 2 + t];
    accA += xv * (ty - train_x[j * 5 + t]);
    accM += xv * ty;
  }
  for (int off = 16; off > 0; off >>= 1) {
    accA += __shfl_down(accA, off, 32);
    accM += __shfl_down(accM, off, 32);
  }
  if (lane == 0) { alpha[t * Nn + row] = accA; msv[t * Nn + row] = accM; }
}

// ---------------------------------------------------------------------------
// Per (task t, 32-row m-tile):
//   stage 1: Ksh[r][n] = os*exp(-.5*sum W (x_r - tx_n)^2)  -> 257KB LDS
//   stage 2: q[r] = k_r^T X k_r via WMMA; A operand from LDS; every X (B)
//            load feeds TWO WMMAs (row groups 0-15 / 16-31)
//   stage 3: k.alpha, k.ms, tx_d*k*ms reductions reusing the LDS K tile
// ---------------------------------------------------------------------------
#define QF_ROWS 32
#define KS_STRIDE 2052     // 2048 + 4: conflict-free, 8B-aligned pairs

__global__ __launch_bounds__(256)
void quadform(const float* __restrict__ mu, const float* __restrict__ action,
              const float* __restrict__ train_x,
              const float* __restrict__ lengthscale,
              const float* __restrict__ outputscale,
              const float* __restrict__ noise,
              const float* __restrict__ Xinv,
              const float* __restrict__ alpha, const float* __restrict__ msv,
              float* __restrict__ varb, float* __restrict__ dka,
              float* __restrict__ s0b, float* __restrict__ s1b)
{
  extern __shared__ float sm[];
  float* Ksh  = sm;                            // QF_ROWS * KS_STRIDE
  float* qsh  = sm + QF_ROWS * KS_STRIDE;      // 32
  float* dksh = qsh + QF_ROWS;                 // 32
  float* s0sh = dksh + QF_ROWS;                // 32
  float* s1sh = s0sh + QF_ROWS;                // 64
  __shared__ float xs[QF_ROWS][5];

  int t   = blockIdx.y;
  int m0  = blockIdx.x * QF_ROWS;
  int tid = threadIdx.x;

  float W[5];
#pragma unroll
  for (int d = 0; d < 5; ++d) {
    float ls = lengthscale[t * 5 + d];
    W[d] = 1.0f / (ls * ls);
  }
  float os = outputscale[t], nz = noise[0];

  if (tid < 32) { qsh[tid] = 0.f; dksh[tid] = 0.f; s0sh[tid] = 0.f; }
  if (tid < 64) s1sh[tid] = 0.f;
  if (tid < QF_ROWS * 5) {
    int r = tid / 5, d = tid % 5, m = m0 + r;
    xs[r][d] = (d < 2) ? mu[m * 2 + d] : action[m * 3 + (d - 2)];
  }
  __syncthreads();

  // ---- stage 1: K tile into LDS
  for (int idx = tid; idx < QF_ROWS * Nn; idx += 256) {
    int r = idx >> 11, n = idx & (Nn - 1);
    float s = 0.f;
#pragma unroll
    for (int d = 0; d < 5; ++d) {
      float df = xs[r][d] - train_x[n * 5 + d];
      s += W[d] * df * df;
    }
    Ksh[r * KS_STRIDE + n] = os * __expf(-0.5f * s);
  }
  __syncthreads();

  // ---- stage 2: q = rowdot(K*X, K) via WMMA (P never materialized)
  int lane = tid & 31, wave = tid >> 5;
  int half = lane >> 4, kk0 = half * 2;
  int cl   = lane & 15;
  const float* Xt = Xinv + (size_t)t * Nn * Nn;

  float q0[8], q1[8];
#pragma unroll
  for (int v = 0; v < 8; ++v) { q0[v] = 0.f; q1[v] = 0.f; }

  for (int jt = wave; jt < Nn / 16; jt += 8) {   // wave-uniform bound
    int c0 = jt * 16;
    v8f acc0 = {}, acc1 = {};
    for (int l = 0; l < Nn; l += 4) {
      v2f b;
      b.x = Xt[(size_t)(l + kk0)     * Nn + c0 + cl];
      b.y = Xt[(size_t)(l + kk0 + 1) * Nn + c0 + cl];
      v2f a0, a1;
      a0.x = Ksh[cl        * KS_STRIDE + l + kk0];
      a0.y = Ksh[cl        * KS_STRIDE + l + kk0 + 1];
      a1.x = Ksh[(16 + cl) * KS_STRIDE + l + kk0];
      a1.y = Ksh[(16 + cl) * KS_STRIDE + l + kk0 + 1];
      acc0 = wmma_f32x4(a0, b, acc0);
      acc1 = wmma_f32x4(a1, b, acc1);
    }
#pragma unroll
    for (int v = 0; v < 8; ++v) {
      int row = v + 8 * half;
      q0[v] += acc0[v] * Ksh[row        * KS_STRIDE + c0 + cl];
      q1[v] += acc1[v] * Ksh[(16 + row) * KS_STRIDE + c0 + cl];
    }
  }
#pragma unroll
  for (int v = 0; v < 8; ++v) {
    atomicAdd(&qsh[v + 8 * half],      q0[v]);
    atomicAdd(&qsh[16 + v + 8 * half], q1[v]);
  }

  // ---- stage 3: cheap reductions reusing the LDS K tile
  for (int r = 0; r < QF_ROWS; ++r) {
    float aA = 0.f, aM = 0.f, a1 = 0.f, a2 = 0.f;
    for (int n = tid; n < Nn; n += 256) {
      float kv = Ksh[r * KS_STRIDE + n];
      float km = kv * msv[t * Nn + n];
      aA += kv * alpha[t * Nn + n];
      aM += km;
      a1 += km * train_x[n * 5 + 0];
      a2 += km * train_x[n * 5 + 1];
    }
    for (int off = 16; off > 0; off >>= 1) {
      aA += __shfl_down(aA, off, 32);
      aM += __shfl_down(aM, off, 32);
      a1 += __shfl_down(a1, off, 32);
      a2 += __shfl_down(a2, off, 32);
    }
    if (lane == 0) {
      atomicAdd(&dksh[r], aA);
      atomicAdd(&s0sh[r], aM);
      atomicAdd(&s1sh[r * 2 + 0], a1);
      atomicAdd(&s1sh[r * 2 + 1], a2);
    }
  }
  __syncthreads();

  if (tid < QF_ROWS) {
    int m = m0 + tid;
    varb[t * Mm + m] = os + nz - qsh[tid];
    dka [t * Mm + m] = dksh[tid];
    s0b [t * Mm + m] = s0sh[tid];
    s1b [(t * Mm + m) * 2 + 0] = s1sh[tid * 2 + 0];
    s1b [(t * Mm + m) * 2 + 1] = s1sh[tid * 2 + 1];
  }
}

// ---------------------------------------------------------------------------
// pred_mu[m][t] = mu[m][t] + (k.alpha)_t
// J[t][d] = -W_td * (x_d * s0_t - s1_td)      (d < 2)
// pred_sigma[m] = diag(var_0, var_1) + J * Sigma_m * J^T
// ---------------------------------------------------------------------------
__global__ __launch_bounds__(256)
void finalize(const float* __restrict__ mu, const float* __restrict__ sigma,
              const float* __restrict__ lengthscale,
              const float* __restrict__ varb, const float* __restrict__ dka,
              const float* __restrict__ s0b, const float* __restrict__ s1b,
              float* __restrict__ out)
{
  int m = blockIdx.x * blockDim.x + threadIdx.x;
  if (m >= Mm) return;
  float x[2] = { mu[m * 2 + 0], mu[m * 2 + 1] };
  float J[2][2];
#pragma unroll
  for (int t = 0; t < 2; ++t) {
    out[m * 2 + t] = x[t] + dka[t * Mm + m];
    float s0 = s0b[t * Mm + m];
#pragma unroll
    for (int d = 0; d < 2; ++d) {
      float ls = lengthscale[t * 5 + d];
      float Wd = 1.0f / (ls * ls);
      J[t][d] = -Wd * (x[d] * s0 - s1b[(t * Mm + m) * 2 + d]);
    }
  }
  float Sg00 = sigma[m * 4 + 0], Sg01 = sigma[m * 4 + 1];
  float Sg10 = sigma[m * 4 + 2], Sg11 = sigma[m * 4 + 3];
  float JS[2][2];
#pragma unroll
  for (int a = 0; a < 2; ++a) {
    JS[a][0] = J[a][0] * Sg00 + J[a][1] * Sg10;
    JS[a][1] = J[a][0] * Sg01 + J[a][1] * Sg11;
  }
  float* ps = out + (size_t)Mm * 2 + (size_t)m * 4;
#pragma unroll
  for (int a = 0; a < 2; ++a)
#pragma unroll
    for (int b = 0; b < 2; ++b) {
      float v = JS[a][0] * J[b][0] + JS[a][1] * J[b][1];
      if (a == b) v += varb[a * Mm + m];
      ps[a * 2 + b] = v;
    }
}

// ---------------------------------------------------------------------------
extern "C" void kernel_launch(void* const* d_in, const int* in_sizes, int n_in,
                              void* d_out, int out_size, void* d_ws, size_t ws_size,
                              hipStream_t stream)
{
  const float* mu          = (const float*)d_in[0];
  const float* sigma       = (const float*)d_in[1];
  const float* action      = (const float*)d_in[2];
  const float* train_x     = (const float*)d_in[3];
  const float* train_y     = (const float*)d_in[4];
  const float* lengthscale = (const float*)d_in[5];
  const float* outputscale = (const float*)d_in[6];
  const float* noise       = (const float*)d_in[7];
  float* out = (float*)d_out;

  // workspace layout (fp32): 4 ping-pong N x N matrices per task + small bufs
  const size_t NNm = (size_t)Nn * Nn;
  float* ws    = (float*)d_ws;
  float* An    = ws;
  float* X     = An + 2 * NNm;
  float* T     = X  + 2 * NNm;
  float* Y     = T  + 2 * NNm;
  float* Sb    = Y  + 2 * NNm;           // 2 (int bits)
  float* alpha = Sb + 2;                 // 2*N
  float* msv   = alpha + 2 * Nn;         // 2*N
  float* varb  = msv   + 2 * Nn;         // 2*M
  float* dka   = varb  + 2 * Mm;         // 2*M
  float* s0b   = dka   + 2 * Mm;         // 2*M
  float* s1b   = s0b   + 2 * Mm;         // 2*M*2
  (void)in_sizes; (void)n_in; (void)out_size; (void)ws_size;

  build_An<<<dim3(Nn / 16, Nn / 16, 2), dim3(16, 16), 0, stream>>>(
      train_x, lengthscale, outputscale, noise, An);
  zeroS<<<1, 32, 0, stream>>>((int*)Sb);
  rowscale<<<dim3(Nn / 256, 2), 256, 0, stream>>>(An, (int*)Sb);
  initX<<<dim3(Nn / 16, Nn / 16, 2), dim3(16, 16), 0, stream>>>((const int*)Sb, X);

  // Newton-Schulz: X <- X*(An*X) + 2X   (An = -A)
  for (int it = 0; it < NS_ITERS; ++it) {
    gemm_ns<<<dim3(Nn / 128, Nn / 128, 2), 256, 0, stream>>>(An, X, An, T, 0.0f);
    gemm_ns<<<dim3(Nn / 128, Nn / 128, 2), 256, 0, stream>>>(X, T, X, Y, 2.0f);
    float* tmp = X; X = Y; Y = tmp;      // even NS_ITERS -> ends in X slot
  }

  solve_vecs<<<dim3(Nn / 8, 2), 256, 0, stream>>>(X, train_x, train_y, alpha, msv);

  size_t qlds = (size_t)(QF_ROWS * KS_STRIDE + 5 * QF_ROWS) * sizeof(float); // ~257.3 KB
  (void)hipFuncSetAttribute((const void*)quadform,
                            hipFuncAttributeMaxDynamicSharedMemorySize, (int)qlds);
  quadform<<<dim3(Mm / QF_ROWS, 2), 256, qlds, stream>>>(
      mu, action, train_x, lengthscale, outputscale, noise,
      X, alpha, msv, varb, dka, s0b, s1b);

  finalize<<<dim3(Mm / 256), 256, 0, stream>>>(
      mu, sigma, lengthscale, varb, dka, s0b, s1b, out);
}